// CosineLoss_17540646437710
// MI455X (gfx1250) — compile-verified
//
#include <hip/hip_runtime.h>
#include <cstdint>

typedef __attribute__((ext_vector_type(2))) float v2f;
typedef __attribute__((ext_vector_type(8))) float v8f;

#define N_ROWS   16384
#define D_COLS   2048
#define TILE_M   16
#define K_CHUNK  128                       // floats of K per double-buffered stage
#define N_CHUNK  (D_COLS / K_CHUNK)        // 16
#define WAVES    4
#define LDS_STRIDE 132                     // 128 + 4 pad dwords: stride%64==4 -> conflict-free
#define TILE_F   (TILE_M * LDS_STRIDE)     // floats per tile buffer (2112)

// One async DMA: 32 lanes x 16B = 512B contiguous global -> LDS (ASYNCcnt path).
__device__ __forceinline__ void async_row_load(uint32_t lds_addr, uint32_t goff,
                                               uint64_t gbase) {
  asm volatile("global_load_async_to_lds_b128 %0, %1, %2"
               :
               : "v"(lds_addr), "v"(goff), "s"(gbase)
               : "memory");
}

__global__ __launch_bounds__(WAVES * 32) void cosine_main(
    const float* __restrict__ cxr, const float* __restrict__ ehr,
    float* __restrict__ accum) {
  __shared__ float lds[WAVES][2][2][TILE_F];  // [wave][buf][matrix][16 x 132]

  const int w = threadIdx.x >> 5;
  const int L = threadIdx.x & 31;
  const int tile = blockIdx.x * WAVES + w;    // 1024 tiles of 16 rows

  const uint64_t aG = (uint64_t)(ehr + (size_t)tile * TILE_M * D_COLS);
  const uint64_t bG = (uint64_t)(cxr + (size_t)tile * TILE_M * D_COLS);

  // LDS byte offsets of the 4 tile buffers (low 32 bits of flat addr == LDS offset)
  uint32_t ldsA[2], ldsB[2];
  ldsA[0] = (uint32_t)(uintptr_t)&lds[w][0][0][0];
  ldsA[1] = (uint32_t)(uintptr_t)&lds[w][1][0][0];
  ldsB[0] = (uint32_t)(uintptr_t)&lds[w][0][1][0];
  ldsB[1] = (uint32_t)(uintptr_t)&lds[w][1][1][0];

  const uint32_t lane_byte = (uint32_t)L * 16u;  // 16B per lane within a 512B row chunk

  auto prefetch = [&](int buf, int c) {
    const uint32_t kbyte = (uint32_t)c * (K_CHUNK * 4);
#pragma unroll
    for (int r = 0; r < TILE_M; ++r) {
      const uint32_t goff = (uint32_t)r * (D_COLS * 4) + lane_byte + kbyte;
      const uint32_t loff = (uint32_t)r * (LDS_STRIDE * 4) + lane_byte;
      async_row_load(ldsA[buf] + loff, goff, aG);
      async_row_load(ldsB[buf] + loff, goff, bG);
    }
  };

  v8f accAB = {};  // diag -> sum a*b
  v8f accAA = {};  // diag -> sum a*a
  v8f accBB = {};  // diag -> sum b*b

  prefetch(0, 0);

  // WMMA 16x16x4 f32 operand address: A and B(=X^T) layouts coincide per lane:
  // row = L&15, k = 2*(L>>4) (+1 in second VGPR) -> one ds_load_b64 per matrix.
  const int base_idx = (L & 15) * LDS_STRIDE + 2 * (L >> 4);

  for (int c = 0; c < N_CHUNK; ++c) {
    const int cur = c & 1;
    if (c + 1 < N_CHUNK) {
      prefetch(cur ^ 1, c + 1);
      // 32 newly-issued outstanding; current chunk (older, in-order) is done at <=32
      asm volatile("s_wait_asynccnt 32" ::: "memory");
    } else {
      asm volatile("s_wait_asynccnt 0" ::: "memory");
    }
    const float* At = &lds[w][cur][0][0];
    const float* Bt = &lds[w][cur][1][0];
#pragma unroll
    for (int s = 0; s < K_CHUNK / 4; ++s) {
      v2f av = *(const v2f*)(At + base_idx + 4 * s);
      v2f bv = *(const v2f*)(Bt + base_idx + 4 * s);
      accAB = __builtin_amdgcn_wmma_f32_16x16x4_f32(
          false, av, false, bv, (short)0, accAB, false, false);
      accAA = __builtin_amdgcn_wmma_f32_16x16x4_f32(
          false, av, false, av, (short)0, accAA, false, false);
      accBB = __builtin_amdgcn_wmma_f32_16x16x4_f32(
          false, bv, false, bv, (short)0, accBB, false, false);
    }
  }

  // Extract diagonal: element (i,i) lives at lane i / vgpr i (i<8)
  // and lane i+16 / vgpr i-8 (i>=8)  [lanes 24..31 hold rows 8..15].
  float ab = 0.f, aa = 0.f, bb = 0.f;
#pragma unroll
  for (int v = 0; v < 8; ++v) {
    if (L == v || L == v + 24) { ab = accAB[v]; aa = accAA[v]; bb = accBB[v]; }
  }
  float cosv = 0.f;
  if (L < 8 || L >= 24) cosv = ab / sqrtf(aa * bb);

  // wave32 reduction of the 16 diag lanes
  for (int o = 16; o > 0; o >>= 1) cosv += __shfl_xor(cosv, o, 32);
  if (L == 0) atomicAdd(accum, cosv);
}

__global__ void zero_accum(float* a) {
  if (threadIdx.x == 0 && blockIdx.x == 0) a[0] = 0.f;
}

__global__ void finalize_loss(const float* a, float* out) {
  if (threadIdx.x == 0 && blockIdx.x == 0)
    out[0] = 1.0f - a[0] * (1.0f / (float)N_ROWS);
}

extern "C" void kernel_launch(void* const* d_in, const int* in_sizes, int n_in,
                              void* d_out, int out_size, void* d_ws, size_t ws_size,
                              hipStream_t stream) {
  const float* cxr = (const float*)d_in[0];
  const float* ehr = (const float*)d_in[1];
  float* accum = (float*)d_ws;
  float* out = (float*)d_out;

  zero_accum<<<1, 32, 0, stream>>>(accum);
  cosine_main<<<(N_ROWS / TILE_M) / WAVES, WAVES * 32, 0, stream>>>(cxr, ehr, accum);
  finalize_loss<<<1, 32, 0, stream>>>(accum, out);
}